// OptimizedMoERouter_83073257439520
// MI455X (gfx1250) — compile-verified
//
#include <hip/hip_runtime.h>
#include <math.h>

typedef __attribute__((ext_vector_type(2))) float v2f;
typedef __attribute__((ext_vector_type(8))) float v8f;

#define D_DIM   2048
#define E_DIM   64
#define TOK_T   64     // tokens per block (4 M-tiles of 16)
#define MT      4      // M tiles per wave
#define NT      4      // N (expert) tiles per wave
#define SLOTS   128    // TOK_T * TOP_K

// ---------------------------------------------------------------------------
// Kernel 1: router GEMM (fp32 WMMA 16x16x4, 4x4 register tiling) + softmax +
// top-2 per 64-token tile. One wave per block; per-block expert histograms and
// prob sums go to workspace so later passes need no atomics.
// ---------------------------------------------------------------------------
__global__ __launch_bounds__(32) void moe_gemm_topk(
    const float* __restrict__ x, const float* __restrict__ W,
    const float* __restrict__ bias,
    int*   __restrict__ ws_idx,    // [N*2]
    float* __restrict__ ws_w,      // [N*2]
    int*   __restrict__ ws_cnt,    // [nblk*64]
    float* __restrict__ ws_ps)     // [nblk*64]
{
    __shared__ float s_logits[TOK_T * E_DIM];   // 16 KB
    __shared__ float s_max[TOK_T];
    __shared__ float s_rden[TOK_T];
    __shared__ int   s_cnt[E_DIM];

    const int lane  = threadIdx.x;            // 0..31
    const int r     = lane & 15;
    const int khalf = (lane >> 4) << 1;       // 0 for lanes 0-15, 2 for 16-31
    const int tok0  = blockIdx.x * TOK_T;

    const float* __restrict__ xr[MT];
    #pragma unroll
    for (int mt = 0; mt < MT; ++mt)
        xr[mt] = x + (size_t)(tok0 + mt * 16 + r) * D_DIM;
    const float* __restrict__ wr[NT];
    #pragma unroll
    for (int t = 0; t < NT; ++t)
        wr[t] = W + (size_t)(t * 16 + r) * D_DIM;

    v8f acc[MT][NT];
    #pragma unroll
    for (int mt = 0; mt < MT; ++mt)
        #pragma unroll
        for (int t = 0; t < NT; ++t)
            acc[mt][t] = (v8f){};

    #pragma unroll 2
    for (int k = 0; k < D_DIM; k += 4) {
        // 32-bit A 16x4 layout: lanes 0-15 hold K=k,k+1 ; lanes 16-31 K=k+2,k+3
        v2f a[MT], b[NT];
        #pragma unroll
        for (int mt = 0; mt < MT; ++mt)
            a[mt] = *(const v2f*)(xr[mt] + k + khalf);
        #pragma unroll
        for (int t = 0; t < NT; ++t)
            b[t] = *(const v2f*)(wr[t] + k + khalf);
        #pragma unroll
        for (int mt = 0; mt < MT; ++mt)
            #pragma unroll
            for (int t = 0; t < NT; ++t)
                acc[mt][t] = __builtin_amdgcn_wmma_f32_16x16x4_f32(
                    false, a[mt], false, b[t], (short)0, acc[mt][t], false, false);
    }

    // C/D layout: VGPR j, lanes 0-15 -> M=j, lanes 16-31 -> M=j+8 ; N = lane&15
    const int mlo = (lane < 16) ? 0 : 8;
    float bv[NT];
    #pragma unroll
    for (int t = 0; t < NT; ++t) bv[t] = bias[t * 16 + r];
    #pragma unroll
    for (int mt = 0; mt < MT; ++mt)
        #pragma unroll
        for (int t = 0; t < NT; ++t)
            #pragma unroll
            for (int j = 0; j < 8; ++j)
                s_logits[(mt * 16 + j + mlo) * E_DIM + t * 16 + r] = acc[mt][t][j] + bv[t];
    s_cnt[lane] = 0;
    s_cnt[lane + 32] = 0;
    __syncthreads();

    // each lane handles tokens (lane) and (lane+32): softmax stats + top-2
    #pragma unroll
    for (int rep = 0; rep < 2; ++rep) {
        const int m = lane + rep * 32;
        float b1v = -3.4e38f, b2v = -3.4e38f;
        int   i1 = 0, i2 = 0;
        for (int e = 0; e < E_DIM; ++e) {
            float v = s_logits[m * E_DIM + e];
            if (v > b1v) { b2v = b1v; i2 = i1; b1v = v; i1 = e; }
            else if (v > b2v) { b2v = v; i2 = e; }
        }
        float den = 0.f;
        for (int e = 0; e < E_DIM; ++e)
            den += expf(s_logits[m * E_DIM + e] - b1v);
        const float rden = 1.0f / den;          // den >= 1 (contains exp(0))
        const float p1 = rden;                  // exp(0)*rden
        const float p2 = expf(b2v - b1v) * rden;
        const float s  = fmaxf(p1 + p2, 1e-6f);
        const int gi = (tok0 + m) * 2;
        ws_idx[gi]     = i1;
        ws_idx[gi + 1] = i2;
        ws_w[gi]       = p1 / s;
        ws_w[gi + 1]   = p2 / s;
        s_max[m]  = b1v;
        s_rden[m] = rden;
        atomicAdd(&s_cnt[i1], 1);
        atomicAdd(&s_cnt[i2], 1);
    }
    __syncthreads();

    // per-block expert histogram
    ws_cnt[blockIdx.x * E_DIM + lane]      = s_cnt[lane];
    ws_cnt[blockIdx.x * E_DIM + lane + 32] = s_cnt[lane + 32];

    // per-block prob sums for importance loss: each lane handles 2 experts
    float ps0 = 0.f, ps1 = 0.f;
    for (int m = 0; m < TOK_T; ++m) {
        const float mx = s_max[m], rd = s_rden[m];
        ps0 += expf(s_logits[m * E_DIM + lane]      - mx) * rd;
        ps1 += expf(s_logits[m * E_DIM + lane + 32] - mx) * rd;
    }
    ws_ps[blockIdx.x * E_DIM + lane]      = ps0;
    ws_ps[blockIdx.x * E_DIM + lane + 32] = ps1;
}

// ---------------------------------------------------------------------------
// Kernel 2: per-expert exclusive scan over block histograms + losses
// ---------------------------------------------------------------------------
__global__ __launch_bounds__(64) void moe_scan_losses(
    const int* __restrict__ ws_cnt, const float* __restrict__ ws_ps,
    int* __restrict__ ws_off, float* __restrict__ d_scalars,
    int nblk, int capacity, int N)
{
    __shared__ float s_kept[E_DIM];
    __shared__ float s_psum[E_DIM];
    const int e = threadIdx.x;

    long long total = 0;
    float ps = 0.f;
    for (int blk = 0; blk < nblk; ++blk) {
        const int idx = blk * E_DIM + e;
        ws_off[idx] = (int)total;
        total += ws_cnt[idx];
        ps += ws_ps[idx];
    }
    s_kept[e] = fminf((float)total, (float)capacity);
    s_psum[e] = ps;
    __syncthreads();

    if (e == 0) {
        const float NK = (float)N * 2.0f;
        const float ideal = 1.0f / (float)E_DIM;
        float lb = 0.f, imp = 0.f, sk = 0.f;
        for (int i = 0; i < E_DIM; ++i) {
            const float t = s_kept[i] / NK;
            lb += (t - ideal) * (t - ideal);
            sk += s_kept[i];
            const float p = s_psum[i] / (float)N;
            imp += (p - ideal) * (p - ideal);
        }
        d_scalars[0] = lb / (float)E_DIM;
        d_scalars[1] = imp / (float)E_DIM;
        d_scalars[2] = 1.0f - sk / NK;
    }
}

// ---------------------------------------------------------------------------
// Kernel 3: capacity mask via (block offset + local rank) in flat order
// ---------------------------------------------------------------------------
__global__ __launch_bounds__(SLOTS) void moe_capacity_mask(
    const int* __restrict__ ws_idx, const float* __restrict__ ws_w,
    const int* __restrict__ ws_off,
    float* __restrict__ out_idx, float* __restrict__ out_w, int capacity)
{
    __shared__ int s_e[SLOTS];
    const int s    = threadIdx.x;                 // slot within group, flat order
    const int base = blockIdx.x * SLOTS;
    const int e    = ws_idx[base + s];
    s_e[s] = e;
    __syncthreads();

    int rank = 0;
    for (int j = 0; j < s; ++j)
        rank += (s_e[j] == e) ? 1 : 0;

    const int pos = ws_off[blockIdx.x * E_DIM + e] + rank;
    const bool keep = pos < capacity;
    out_idx[base + s] = keep ? (float)e : -1.0f;
    out_w[base + s]   = keep ? ws_w[base + s] : 0.0f;
}

// ---------------------------------------------------------------------------
extern "C" void kernel_launch(void* const* d_in, const int* in_sizes, int n_in,
                              void* d_out, int out_size, void* d_ws, size_t ws_size,
                              hipStream_t stream) {
    const float* x = (const float*)d_in[0];
    const float* W = (const float*)d_in[1];
    const float* b = (const float*)d_in[2];

    const int E = in_sizes[2];                 // 64
    const int D = in_sizes[1] / E;             // 2048
    const int N = in_sizes[0] / D;             // 32768 tokens
    const int nblk = N / TOK_T;                // 512
    const int capacity = (int)((5LL * N * 2) / (4LL * E));   // int(1.25*N*K/E)

    // workspace carve-up
    int*   ws_idx = (int*)d_ws;                     // N*2
    float* ws_w   = (float*)(ws_idx + (size_t)N * 2);
    int*   ws_cnt = (int*)(ws_w + (size_t)N * 2);   // nblk*64
    int*   ws_off = ws_cnt + (size_t)nblk * E_DIM;
    float* ws_ps  = (float*)(ws_off + (size_t)nblk * E_DIM);

    float* out_idx     = (float*)d_out;
    float* out_w       = out_idx + (size_t)N * 2;
    float* out_scalars = out_w + (size_t)N * 2;

    moe_gemm_topk<<<nblk, 32, 0, stream>>>(x, W, b, ws_idx, ws_w, ws_cnt, ws_ps);
    moe_scan_losses<<<1, 64, 0, stream>>>(ws_cnt, ws_ps, ws_off, out_scalars,
                                          nblk, capacity, N);
    moe_capacity_mask<<<nblk, SLOTS, 0, stream>>>(ws_idx, ws_w, ws_off,
                                                  out_idx, out_w, capacity);
}